// GeneticAlgorithm_78460462563868
// MI455X (gfx1250) — compile-verified
//
#include <hip/hip_runtime.h>
#include <math.h>

// GA generation step, MI455X (gfx1250, wave32).
// Bandwidth-bound: ~0.84 GB mandatory traffic -> ~36us at 23.3 TB/s.
// One workgroup per row (4096 rows), 256 threads (8 wave32s).
// pop row staged in registers (16 float4/thread, read from HBM exactly once);
// u_mask row staged into LDS via gfx1250 async global->LDS copy
// (global_load_async_to_lds_b128 / ASYNCcnt), overlapped with the pop pass;
// noise loaded only where mask < rate (EXEC-masked, ~27% of lines);
// NT store on the output stream.

typedef float v4f __attribute__((ext_vector_type(4)));

#define MUT_RATE 0.01f

#define ASL __attribute__((address_space(3)))

// Use the hand-rolled async path (ISA-doc semantics, fails loudly at assembly
// time if the syntax is wrong — unlike a builtin with guessed operand order).
#define USE_ASYNC_LDS 1

#if defined(__has_builtin) && __has_builtin(__builtin_amdgcn_s_wait_asynccnt)
#  define WAIT_ASYNC0() __builtin_amdgcn_s_wait_asynccnt(0)
#else
#  define WAIT_ASYNC0() asm volatile("s_wait_asynccnt 0x0" ::: "memory")
#endif

// GLOBAL_LOAD_ASYNC_TO_LDS_B128, GV mode (ISA §15.18 / §10.7):
//   each lane copies 16 bytes from its global address to its LDS offset (ASYNCcnt)
__device__ __forceinline__ void async_g2l_b128(unsigned lds_off, const void* gptr)
{
    asm volatile("global_load_async_to_lds_b128 %0, %1, off"
                 :
                 : "v"(lds_off), "v"(gptr)
                 : "memory");
}

// ---------------- fast path: L == 16384, 256 threads, 16 chunks ----------------
template <int CHUNKS>
__launch_bounds__(256)
__global__ void ga_fast_kernel(const float* __restrict__ pop,
                               const int*   __restrict__ start_idx,
                               const float* __restrict__ u_mask,
                               const float* __restrict__ u_noise,
                               const int*   __restrict__ seg_len_p,
                               float*       __restrict__ out,
                               int L, int half)
{
    const int row  = blockIdx.x;
    const int tid  = threadIdx.x;
    const int lane = tid & 31;
    const int wave = tid >> 5;

    const int s    = start_idx[(row < half) ? row : (row - half)];
    const int seg  = *seg_len_p;                       // 8192
    const int pair = (row < half) ? (row + half) : (row - half);
    const int ob   = row  * L;                         // < 2^27, fits int
    const int pb   = pair * L;

#if USE_ASYNC_LDS
    // Stage this row's mask (64 KB) into LDS asynchronously; completes under
    // the phase-1 pop streaming + reduction. Each thread copies exactly the
    // chunks it will consume in phase 2, so s_wait_asynccnt 0 suffices (the
    // wave's own ASYNCcnt hitting 0 means its LDS data has landed).
    __shared__ float lds_mask[CHUNKS * 256 * 4];
    #pragma unroll
    for (int i = 0; i < CHUNKS; ++i) {
        const int c = 4 * (i * 256 + tid);
        const unsigned lds_off =
            (unsigned)(unsigned long long)(ASL float*)(lds_mask + c);
        async_g2l_b128(lds_off, (const void*)(u_mask + ob + c));
    }
#else
    // Fallback: prefetch the row's mask into cache while phase 1 streams pop.
    {
        const char* mb = (const char*)(u_mask + ob);
        __builtin_prefetch(mb + (size_t)tid * 128u, 0, 0);
        __builtin_prefetch(mb + ((size_t)tid + 256u) * 128u, 0, 0);
    }
#endif

    // ---- Phase 1: load crossed row into registers, accumulate stats ----
    v4f  v[CHUNKS];
    float sum = 0.0f, mx = -INFINITY, mn = INFINITY;

    #pragma unroll
    for (int i = 0; i < CHUNKS; ++i) {
        const int  c   = 4 * (i * 256 + tid);
        const bool in0 = (c     >= s) && (c     < s + seg);
        const bool in3 = (c + 3 >= s) && (c + 3 < s + seg);
        v4f t;
        if (in0 == in3) {
            // Uniform chunk (valid since seg_len >= 4): single B128 load.
            t = *(const v4f*)(pop + (in0 ? pb : ob) + c);
        } else {
            // Chunk straddles a crossover boundary: per-component select.
            #pragma unroll
            for (int k = 0; k < 4; ++k) {
                const int  ck = c + k;
                const bool sw = (ck >= s) && (ck < s + seg);
                t[k] = pop[(sw ? pb : ob) + ck];
            }
        }
        v[i] = t;
        sum += (t.x + t.y) + (t.z + t.w);
        mx = fmaxf(mx, fmaxf(fmaxf(t.x, t.y), fmaxf(t.z, t.w)));
        mn = fminf(mn, fminf(fminf(t.x, t.y), fminf(t.z, t.w)));
    }

    // ---- Wave32 butterfly reduction ----
    #pragma unroll
    for (int off = 16; off > 0; off >>= 1) {
        sum += __shfl_xor(sum, off, 32);
        mx = fmaxf(mx, __shfl_xor(mx, off, 32));
        mn = fminf(mn, __shfl_xor(mn, off, 32));
    }

    __shared__ float s_sum[8], s_mx[8], s_mn[8];
    if (lane == 0) { s_sum[wave] = sum; s_mx[wave] = mx; s_mn[wave] = mn; }
    __syncthreads();

    float tsum = 0.0f, tmx = -INFINITY, tmn = INFINITY;
    #pragma unroll
    for (int w = 0; w < 8; ++w) {
        tsum += s_sum[w];
        tmx = fmaxf(tmx, s_mx[w]);
        tmn = fminf(tmn, s_mn[w]);
    }
    const float avg = tsum / (float)L;

#if USE_ASYNC_LDS
    WAIT_ASYNC0();   // this wave's mask chunks are resident in LDS
#endif

    // ---- Phase 2: mutate + select, stream out ----
    #pragma unroll
    for (int i = 0; i < CHUNKS; ++i) {
        const int c   = 4 * (i * 256 + tid);
        const int idx = ob + c;
#if USE_ASYNC_LDS
        const v4f m   = *(const v4f*)(lds_mask + c);
#else
        const v4f m   = __builtin_nontemporal_load((const v4f*)(u_mask + idx));
#endif
        const v4f t   = v[i];
        v4f o = t;
        #pragma unroll
        for (int k = 0; k < 4; ++k) {
            if (m[k] < MUT_RATE) {                 // ~1%: EXEC-masked noise load
                const float nz = u_noise[idx + k];
                o[k] = fminf(fmaxf(t[k] + nz * avg, tmn), tmx);
            }
        }
        __builtin_nontemporal_store(o, (v4f*)(out + idx));
    }
}

// ---------------- generic fallback (any L, any block size) ----------------
__global__ void ga_generic_kernel(const float* __restrict__ pop,
                                  const int*   __restrict__ start_idx,
                                  const float* __restrict__ u_mask,
                                  const float* __restrict__ u_noise,
                                  const int*   __restrict__ seg_len_p,
                                  float*       __restrict__ out,
                                  int L, int half)
{
    const int row  = blockIdx.x;
    const int pair = (row < half) ? (row + half) : (row - half);
    const int s    = start_idx[(row < half) ? row : (row - half)];
    const int seg  = *seg_len_p;
    const long ob  = (long)row  * L;
    const long pb  = (long)pair * L;

    float sum = 0.0f, mx = -INFINITY, mn = INFINITY;
    for (int c = threadIdx.x; c < L; c += blockDim.x) {
        const bool sw = (c >= s) && (c < s + seg);
        const float x = pop[(sw ? pb : ob) + c];
        sum += x; mx = fmaxf(mx, x); mn = fminf(mn, x);
    }
    for (int off = 16; off > 0; off >>= 1) {
        sum += __shfl_xor(sum, off, 32);
        mx = fmaxf(mx, __shfl_xor(mx, off, 32));
        mn = fminf(mn, __shfl_xor(mn, off, 32));
    }
    __shared__ float s_sum[32], s_mx[32], s_mn[32];
    const int lane = threadIdx.x & 31, wave = threadIdx.x >> 5;
    const int nw   = (blockDim.x + 31) >> 5;
    if (lane == 0) { s_sum[wave] = sum; s_mx[wave] = mx; s_mn[wave] = mn; }
    __syncthreads();
    float tsum = 0.0f, tmx = -INFINITY, tmn = INFINITY;
    for (int w = 0; w < nw; ++w) {
        tsum += s_sum[w]; tmx = fmaxf(tmx, s_mx[w]); tmn = fminf(tmn, s_mn[w]);
    }
    const float avg = tsum / (float)L;

    for (int c = threadIdx.x; c < L; c += blockDim.x) {
        const bool sw = (c >= s) && (c < s + seg);
        const long idx = ob + c;
        const float x = pop[(sw ? pb : ob) + c];
        float o = x;
        if (u_mask[idx] < MUT_RATE)
            o = fminf(fmaxf(x + u_noise[idx] * avg, tmn), tmx);
        out[idx] = o;
    }
}

extern "C" void kernel_launch(void* const* d_in, const int* in_sizes, int n_in,
                              void* d_out, int out_size, void* d_ws, size_t ws_size,
                              hipStream_t stream)
{
    const float* pop       = (const float*)d_in[0];
    const int*   start_idx = (const int*)  d_in[1];
    const float* u_mask    = (const float*)d_in[2];
    const float* u_noise   = (const float*)d_in[3];
    const int*   seg_len   = (const int*)  d_in[4];
    float*       out       = (float*)d_out;

    const int half = in_sizes[1];          // 2048
    const int N    = half * 2;             // 4096
    const int L    = in_sizes[0] / N;      // 16384

    if (L == 16384) {
        ga_fast_kernel<16><<<N, 256, 0, stream>>>(pop, start_idx, u_mask, u_noise,
                                                  seg_len, out, L, half);
    } else {
        ga_generic_kernel<<<N, 256, 0, stream>>>(pop, start_idx, u_mask, u_noise,
                                                 seg_len, out, L, half);
    }
}